// CaMoE_Block_45457933861039
// MI455X (gfx1250) — compile-verified
//
#include <hip/hip_runtime.h>
#include <hip/hip_bf16.h>
#include <math.h>

#define NTOK 4096          // B*T = 2*2048
#define CDIM 1024
#define HDIM 4096
#define NEXP 8
#define NRWKV 6

#define USE_ASYNC 1

typedef __attribute__((ext_vector_type(2))) float v2f;
typedef __attribute__((ext_vector_type(8))) float v8f;

enum { EPI_NONE = 0, EPI_RELU2 = 1, EPI_GELU = 2, EPI_ADD = 3, EPI_GATED = 4 };

__device__ __forceinline__ float gelu_tanh(float x) {
  const float k0 = 0.7978845608028654f, k1 = 0.044715f;
  return 0.5f * x * (1.0f + tanhf(k0 * (x + k1 * x * x * x)));
}

__device__ __forceinline__ void async_ld_b128(unsigned ldsOff, const float* g) {
#if USE_ASYNC
  asm volatile("global_load_async_to_lds_b128 %0, %1, off"
               :: "v"(ldsOff), "v"(g) : "memory");
#endif
}
__device__ __forceinline__ void wait_async() {
#if USE_ASYNC
  asm volatile("s_wait_asynccnt 0x0" ::: "memory");
#endif
}

// ---------------------------------------------------------------------------
// fp32 WMMA GEMM: Out[M x Nn] = epi(A[M x K] @ B[K x Nn])
// Block = 256 threads (8 waves). Tile 64(M) x 128(N); wave grid 2x4, each wave
// computes a 32x32 patch as 2x2 WMMA 16x16 tiles. K stepped by 16.
// A[64x16] and B[16x128] staged in LDS, double-buffered; full tiles use
// global_load_async_to_lds_b128 (ASYNCcnt), tail tiles use guarded loads.
// Optional: gather A rows through rowList (entries token*2+slot), dynamic M
// from *rowCount, gated scatter into per-slot contribution buffers.
// ---------------------------------------------------------------------------
template <int EPI, bool GATHER_A>
__global__ __launch_bounds__(256) void gemm_f32_wmma(
    const float* __restrict__ A, int lda,
    const float* __restrict__ B, int ldb,
    float* __restrict__ Out, int ldo,
    const float* __restrict__ AddSrc,
    const int* __restrict__ rowList,
    const int* __restrict__ rowCount,
    int M, int K,
    const float* __restrict__ gates,
    float* __restrict__ contrib)
{
  const int tid  = threadIdx.x;
  const int lane = tid & 31;
  const int wave = tid >> 5;
  const int wm   = wave >> 2;   // 0..1
  const int wn   = wave & 3;    // 0..3
  const int rowTile = blockIdx.y * 64;
  const int colBlk  = blockIdx.x * 128;
  const int effM = rowCount ? *rowCount : M;
  if (rowTile >= effM) return;  // whole block exits before any barrier
  const bool fullTile = (rowTile + 64 <= effM);

  // Row strides padded so LDS b128 stores stay 16B aligned and fragment reads
  // are bank-conflict free (20 and 144 dwords).
  __shared__ float Ald[2][64][20];
  __shared__ float Bld[2][16][144];

  // -------- staging assignments (fixed per thread) --------
  const int ar = tid >> 2;            // 0..63 : A row within tile
  const int ac = (tid & 3) * 4;       // A col segment (float4)
  const int aGr = rowTile + ar;
  const bool aValid = aGr < effM;
  long aSrcRow = 0;
  if (aValid) aSrcRow = GATHER_A ? (long)(rowList[aGr] >> 1) : (long)aGr;
  const float* aPtr = A + aSrcRow * lda + ac;

  const int bc  = lane * 4;           // 0..124 : B col segment (float4)
  const int bk0 = tid >> 5;           // 0..7   : B row (and +8)
  const float* bPtr0 = B + (long)bk0 * ldb + colBlk + bc;
  const float* bPtr1 = B + (long)(bk0 + 8) * ldb + colBlk + bc;

  unsigned aLds[2], bLds0[2], bLds1[2];
#pragma unroll
  for (int b = 0; b < 2; ++b) {
    aLds[b]  = (unsigned)(uintptr_t)(void*)&Ald[b][ar][ac];
    bLds0[b] = (unsigned)(uintptr_t)(void*)&Bld[b][bk0][bc];
    bLds1[b] = (unsigned)(uintptr_t)(void*)&Bld[b][bk0 + 8][bc];
  }

  auto stage = [&](int b, int kk) {
    if (fullTile) {
#if USE_ASYNC
      async_ld_b128(aLds[b], aPtr + kk);
      async_ld_b128(bLds0[b], bPtr0 + (long)kk * ldb);
      async_ld_b128(bLds1[b], bPtr1 + (long)kk * ldb);
#else
      *(float4*)&Ald[b][ar][ac] = *(const float4*)(aPtr + kk);
      *(float4*)&Bld[b][bk0][bc] = *(const float4*)(bPtr0 + (long)kk * ldb);
      *(float4*)&Bld[b][bk0 + 8][bc] = *(const float4*)(bPtr1 + (long)kk * ldb);
#endif
    } else {
      float4 av = make_float4(0.f, 0.f, 0.f, 0.f);
      if (aValid) av = *(const float4*)(aPtr + kk);
      *(float4*)&Ald[b][ar][ac] = av;
      *(float4*)&Bld[b][bk0][bc] = *(const float4*)(bPtr0 + (long)kk * ldb);
      *(float4*)&Bld[b][bk0 + 8][bc] = *(const float4*)(bPtr1 + (long)kk * ldb);
    }
  };

  const int l15   = lane & 15;
  const int khalf = lane >> 4;  // 0: K pair {0,1}, 1: K pair {2,3}
  const int mrow0 = wm * 32 + l15;
  const int ncol0 = wn * 32 + l15;

  v8f acc[2][2] = {};

  auto compute = [&](int b) {
#pragma unroll
    for (int s = 0; s < 4; ++s) {
      v2f a[2], bb[2];
#pragma unroll
      for (int mi = 0; mi < 2; ++mi) {
        a[mi].x = Ald[b][mrow0 + mi * 16][4 * s + 2 * khalf];
        a[mi].y = Ald[b][mrow0 + mi * 16][4 * s + 2 * khalf + 1];
      }
#pragma unroll
      for (int ni = 0; ni < 2; ++ni) {
        bb[ni].x = Bld[b][4 * s + khalf][ncol0 + ni * 16];
        bb[ni].y = Bld[b][4 * s + 2 + khalf][ncol0 + ni * 16];
      }
#pragma unroll
      for (int mi = 0; mi < 2; ++mi)
#pragma unroll
        for (int ni = 0; ni < 2; ++ni)
          acc[mi][ni] = __builtin_amdgcn_wmma_f32_16x16x4_f32(
              false, a[mi], false, bb[ni], (short)0, acc[mi][ni], false, false);
    }
  };

  // -------- main loop: double-buffered K steps of 16 --------
  int buf = 0;
  stage(0, 0);
  wait_async();
  __syncthreads();
  for (int kk = 0; kk < K; kk += 16) {
    if (kk + 16 < K) stage(buf ^ 1, kk + 16);
    compute(buf);
    wait_async();
    __syncthreads();
    buf ^= 1;
  }

  // -------- epilogue --------
#pragma unroll
  for (int mi = 0; mi < 2; ++mi) {
#pragma unroll
    for (int ni = 0; ni < 2; ++ni) {
#pragma unroll
      for (int v = 0; v < 8; ++v) {
        int rloc = wm * 32 + mi * 16 + ((lane < 16) ? v : (v + 8));
        int gr = rowTile + rloc;
        if (gr >= effM) continue;
        int col = colBlk + wn * 32 + ni * 16 + l15;
        float val = acc[mi][ni][v];
        if (EPI == EPI_RELU2) { float t = fmaxf(val, 0.0f); val = t * t; }
        else if (EPI == EPI_GELU) { val = gelu_tanh(val); }
        else if (EPI == EPI_ADD) { val += AddSrc[(long)gr * ldo + col]; }
        if (EPI == EPI_GATED) {
          int entry = rowList[gr];
          int tok = entry >> 1, slot = entry & 1;
          float g = gates[tok * 2 + slot];
          contrib[((long)slot * NTOK + tok) * CDIM + col] = val * g;
        } else {
          Out[(long)gr * ldo + col] = val;
        }
      }
    }
  }
}

// ---------------------------------------------------------------------------
// LayerNorm (optionally fused residual add; optionally emits the pre-LN sum)
// ---------------------------------------------------------------------------
__global__ __launch_bounds__(256) void ln_kernel(
    const float* __restrict__ x, const float* __restrict__ addend,
    const float* __restrict__ g, const float* __restrict__ b,
    float* __restrict__ sumOut, float* __restrict__ out)
{
  const int row = blockIdx.x;
  const int tid = threadIdx.x;
  __shared__ float rs[256], rs2[256];
  float vals[CDIM / 256];
  float s = 0.f, s2 = 0.f;
#pragma unroll
  for (int i = 0; i < CDIM / 256; ++i) {
    int c = tid + 256 * i;
    float v = x[(long)row * CDIM + c];
    if (addend) v += addend[(long)row * CDIM + c];
    vals[i] = v;
    s += v;
    s2 += v * v;
  }
  rs[tid] = s; rs2[tid] = s2;
  __syncthreads();
  for (int st = 128; st > 0; st >>= 1) {
    if (tid < st) { rs[tid] += rs[tid + st]; rs2[tid] += rs2[tid + st]; }
    __syncthreads();
  }
  float mean = rs[0] * (1.0f / CDIM);
  float var  = rs2[0] * (1.0f / CDIM) - mean * mean;
  float inv  = rsqrtf(var + 1e-5f);
#pragma unroll
  for (int i = 0; i < CDIM / 256; ++i) {
    int c = tid + 256 * i;
    float v = vals[i];
    if (sumOut) sumOut[(long)row * CDIM + c] = v;
    out[(long)row * CDIM + c] = (v - mean) * inv * g[c] + b[c];
  }
}

// ---------------------------------------------------------------------------
// Router + sparse dispatch list construction. One block (256 thr) per token.
// ---------------------------------------------------------------------------
__global__ __launch_bounds__(256) void router_kernel(
    const float* __restrict__ h,
    const float* __restrict__ w_conf,   // [E,C]
    const float* __restrict__ w_diff,   // [C]
    const float* __restrict__ W_aff,    // [C,E]
    const float* __restrict__ cap,      // [E]
    float* __restrict__ winners_out,    // [N,2] (as float)
    float* __restrict__ costs_out,      // [N]
    float* __restrict__ diff_out,       // [N]
    float* __restrict__ aff_out,        // [N,E]
    float* __restrict__ gates,          // [N,2]
    int* __restrict__ lists,            // [E,N]
    int* __restrict__ cnt)              // [E]
{
  const int tok = blockIdx.x;
  const int tid = threadIdx.x;
  float hv[CDIM / 256];
#pragma unroll
  for (int i = 0; i < CDIM / 256; ++i)
    hv[i] = h[(long)tok * CDIM + tid + 256 * i];

  __shared__ float red[256];
  __shared__ float dots[17];
  for (int j = 0; j < 17; ++j) {
    float p = 0.f;
#pragma unroll
    for (int i = 0; i < CDIM / 256; ++i) {
      int c = tid + 256 * i;
      float w;
      if (j < 8)       w = w_conf[j * CDIM + c];
      else if (j == 8) w = w_diff[c];
      else             w = W_aff[(long)c * NEXP + (j - 9)];
      p += hv[i] * w;
    }
    red[tid] = p;
    __syncthreads();
    for (int st = 128; st > 0; st >>= 1) {
      if (tid < st) red[tid] += red[tid + st];
      __syncthreads();
    }
    if (tid == 0) dots[j] = red[0];
    __syncthreads();
  }

  if (tid == 0) {
    float bids[NEXP];
    float d = dots[8];
    float diff = (d > 20.f) ? d : log1pf(expf(d));   // softplus
    for (int e = 0; e < NEXP; ++e) {
      float conf = 1.0f / (1.0f + expf(-dots[e]));   // sigmoid
      float aff  = dots[9 + e];
      bids[e] = conf * cap[e] + aff;
      aff_out[(long)tok * NEXP + e] = aff;
    }
    int i0 = 0;
    for (int e = 1; e < NEXP; ++e) if (bids[e] > bids[i0]) i0 = e;
    int i1 = (i0 == 0) ? 1 : 0;
    for (int e = 0; e < NEXP; ++e) {
      if (e == i0 || e == i1) continue;
      if (bids[e] > bids[i1]) i1 = e;
    }
    float v0 = bids[i0], v1 = bids[i1];
    float e1 = expf(v1 - v0);
    float w0 = 1.0f / (1.0f + e1);
    float w1 = e1 / (1.0f + e1);
    winners_out[tok * 2 + 0] = (float)i0;
    winners_out[tok * 2 + 1] = (float)i1;
    costs_out[tok] = (v0 * w0 + v1 * w1) * diff;
    diff_out[tok] = diff;
    gates[tok * 2 + 0] = w0;
    gates[tok * 2 + 1] = w1;
    int p0 = atomicAdd(&cnt[i0], 1); lists[i0 * NTOK + p0] = tok * 2;
    int p1 = atomicAdd(&cnt[i1], 1); lists[i1 * NTOK + p1] = tok * 2 + 1;
  }
}

__global__ __launch_bounds__(256) void add_kernel(
    const float* __restrict__ a, const float* __restrict__ b,
    float* __restrict__ o, long n)
{
  long i = (long)blockIdx.x * 256 + threadIdx.x;
  if (i < n) o[i] = a[i] + b[i];
}

__global__ __launch_bounds__(256) void combine_kernel(
    float* __restrict__ xo, const float* __restrict__ c0,
    const float* __restrict__ c1, long n)
{
  long i = (long)blockIdx.x * 256 + threadIdx.x;
  if (i < n) xo[i] = xo[i] + c0[i] + c1[i];
}

extern "C" void kernel_launch(void* const* d_in, const int* in_sizes, int n_in,
                              void* d_out, int out_size, void* d_ws, size_t ws_size,
                              hipStream_t stream) {
  (void)in_sizes; (void)n_in; (void)out_size; (void)ws_size;
  const float* x       = (const float*)d_in[0];
  const float* v_first = (const float*)d_in[1];
  const float* cap     = (const float*)d_in[2];
  const float* ln1_g   = (const float*)d_in[3];
  const float* ln1_b   = (const float*)d_in[4];
  const float* ln2_g   = (const float*)d_in[5];
  const float* ln2_b   = (const float*)d_in[6];
  const float* Wa      = (const float*)d_in[7];
  const float* Ws      = (const float*)d_in[8];
  const float* w_conf  = (const float*)d_in[9];
  const float* w_diff  = (const float*)d_in[10];
  const float* W_aff   = (const float*)d_in[11];
  const float* Wb_h    = (const float*)d_in[12];
  const float* Wb_s    = (const float*)d_in[13];
  const float* Wk_r    = (const float*)d_in[14];
  const float* Wv_r    = (const float*)d_in[15];
  const float* W1_t    = (const float*)d_in[16];
  const float* W2_t    = (const float*)d_in[17];

  const long NC = (long)NTOK * CDIM;

  float* out       = (float*)d_out;
  float* x_out     = out;                 // [N,C]  (holds x2 until combine)
  float* vf_out    = out + NC;            // [N,C]
  float* win_out   = out + 2 * NC;        // [N,2]
  float* costs_out = win_out + NTOK * 2;  // [N]
  float* diffo     = costs_out + NTOK;    // [N]
  float* affo      = diffo + NTOK;        // [N,E]

  float* w    = (float*)d_ws;
  float* x_ln = w;  w += NC;
  float* att  = w;  w += NC;
  float* st   = w;  w += NC;
  float* h    = w;  w += NC;
  float* t1   = w;  w += NC;              // prefix accumulator
  float* z    = w;  w += NC;
  float* kbuf = w;  w += (long)NTOK * HDIM;  // expert hidden (reused)
  float* ctb  = w;  w += 2 * NC;          // per-slot contributions
  float* gates = w; w += NTOK * 2;
  int* lists = (int*)w;
  int* cnt   = lists + NEXP * NTOK;

  hipMemsetAsync(cnt, 0, NEXP * sizeof(int), stream);

  const dim3 blk(256);
  const dim3 gC(CDIM / 128, NTOK / 64);   // Nn=1024 GEMMs
  const dim3 gH(HDIM / 128, NTOK / 64);   // Nn=4096 GEMMs

  // --- TimeMix stand-in ---
  ln_kernel<<<NTOK, 256, 0, stream>>>(x, nullptr, ln1_g, ln1_b, nullptr, x_ln);
  gemm_f32_wmma<EPI_NONE, false><<<gC, blk, 0, stream>>>(
      x_ln, CDIM, Wa, CDIM, att, CDIM, nullptr, nullptr, nullptr, NTOK, CDIM,
      nullptr, nullptr);
  gemm_f32_wmma<EPI_NONE, false><<<gC, blk, 0, stream>>>(
      x_ln, CDIM, Ws, CDIM, st, CDIM, nullptr, nullptr, nullptr, NTOK, CDIM,
      nullptr, nullptr);
  // x2 = x + att (stored into x_out), h = LN2(x2)
  ln_kernel<<<NTOK, 256, 0, stream>>>(x, att, ln2_g, ln2_b, x_out, h);

  // --- router (+ builds sparse dispatch lists) ---
  router_kernel<<<NTOK, 256, 0, stream>>>(h, w_conf, w_diff, W_aff, cap,
                                          win_out, costs_out, diffo, affo,
                                          gates, lists, cnt);

  // --- bridge prefix: t1 = h@Wb_h; t1 += state@Wb_s; z = h + t1 ---
  gemm_f32_wmma<EPI_NONE, false><<<gC, blk, 0, stream>>>(
      h, CDIM, Wb_h, CDIM, t1, CDIM, nullptr, nullptr, nullptr, NTOK, CDIM,
      nullptr, nullptr);
  gemm_f32_wmma<EPI_ADD, false><<<gC, blk, 0, stream>>>(
      st, CDIM, Wb_s, CDIM, t1, CDIM, t1, nullptr, nullptr, NTOK, CDIM,
      nullptr, nullptr);
  add_kernel<<<NC / 256, 256, 0, stream>>>(h, t1, z, NC);

  // --- RWKV FFN experts (sparse, gathered): relu(h Wk)^2 Wv ---
  for (int e = 0; e < NRWKV; ++e) {
    gemm_f32_wmma<EPI_RELU2, true><<<gH, blk, 0, stream>>>(
        h, CDIM, Wk_r + (long)e * CDIM * HDIM, HDIM, kbuf, HDIM, nullptr,
        lists + e * NTOK, cnt + e, NTOK, CDIM, nullptr, nullptr);
    gemm_f32_wmma<EPI_GATED, false><<<gC, blk, 0, stream>>>(
        kbuf, HDIM, Wv_r + (long)e * HDIM * CDIM, CDIM, nullptr, CDIM, nullptr,
        lists + e * NTOK, cnt + e, NTOK, HDIM, gates, ctb);
  }

  // --- transformer experts (sparse, gathered): gelu(z W1) W2 ---
  for (int e = 0; e < 2; ++e) {
    gemm_f32_wmma<EPI_GELU, true><<<gC, blk, 0, stream>>>(
        z, CDIM, W1_t + (long)e * CDIM * CDIM, CDIM, kbuf, CDIM, nullptr,
        lists + (NRWKV + e) * NTOK, cnt + NRWKV + e, NTOK, CDIM, nullptr,
        nullptr);
    gemm_f32_wmma<EPI_GATED, false><<<gC, blk, 0, stream>>>(
        kbuf, CDIM, W2_t + (long)e * CDIM * CDIM, CDIM, nullptr, CDIM, nullptr,
        lists + (NRWKV + e) * NTOK, cnt + NRWKV + e, NTOK, CDIM, gates, ctb);
  }

  // --- x_out = x2 + slot0 + slot1 contributions; v_first passthrough ---
  combine_kernel<<<NC / 256, 256, 0, stream>>>(x_out, ctb, ctb + NC, NC);
  hipMemcpyAsync(vf_out, v_first, NC * sizeof(float),
                 hipMemcpyDeviceToDevice, stream);
}